// RoIPool_37950331027657
// MI455X (gfx1250) — compile-verified
//
#include <hip/hip_runtime.h>
#include <hip/hip_bf16.h>

// RoI max-pool for MI455X (gfx1250).
//
// Roofline: input 5.1 MB (fits trivially in 192 MB L2), output 12.8 MB,
// ~4 reads per output -> purely memory/launch bound (few microseconds at
// 23.3 TB/s). No inner products anywhere -> WMMA does not apply. The CDNA5
// feature that *does* apply is the Tensor Data Mover: each RoI window is a
// strided 3D tile (winW x winH x channels, strides W and H*W) reused by all
// 49 bins, so one TENSOR_LOAD_TO_LDS per (roi, channel-chunk) block replaces
// thousands of per-lane loads + address math. Fallback: per-lane
// GLOBAL_LOAD_ASYNC_TO_LDS_B32 (ASYNCcnt) if the TDM builtin is absent.

#define POOL 7
#define CCHUNK 32
#define CELL_CAP 512                       // max window cells per channel
#define SMEM_FLOATS (CCHUNK * CELL_CAP)    // 16384 floats = 64 KB LDS

#if defined(__has_builtin)
#  if __has_builtin(__builtin_amdgcn_tensor_load_to_lds)
#    define HAVE_TDM 1
#  endif
#endif

typedef unsigned int u32x4 __attribute__((ext_vector_type(4)));
typedef int          i32x8 __attribute__((ext_vector_type(8)));
typedef int          i32x4 __attribute__((ext_vector_type(4)));

// Fallback path: GLOBAL_LOAD_ASYNC_TO_LDS_B32, tracked by ASYNCcnt.
__device__ __forceinline__ void async_ld_f32_to_lds(const float* g, unsigned lds_byte_addr) {
    asm volatile("global_load_async_to_lds_b32 %0, %1, off"
                 :: "v"(lds_byte_addr), "v"(g)
                 : "memory");
}
__device__ __forceinline__ void wait_asynccnt0() {
    asm volatile("s_wait_asynccnt 0x0" ::: "memory");
}

__device__ __forceinline__ float clipf(float v, float lo, float hi) {
    return fminf(fmaxf(v, lo), hi);
}

__global__ __launch_bounds__(256)
void roipool_kernel(const float* __restrict__ x, const float* __restrict__ rois,
                    float* __restrict__ out,
                    int C, int H, int W, float scale) {
    __shared__ float sm[SMEM_FLOATS];

    const int k   = blockIdx.x;          // roi index
    const int c0  = blockIdx.y * CCHUNK; // channel chunk base
    const int tid = threadIdx.x;
    const int cc  = min(CCHUNK, C - c0);

    // ---- RoI geometry, mirroring the reference's fp32 math exactly ----
    const float r0 = rois[k * 5 + 0];
    const float x1 = rintf(rois[k * 5 + 1] * scale);  // rintf == jnp.round (RNE)
    const float y1 = rintf(rois[k * 5 + 2] * scale);
    const float x2 = rintf(rois[k * 5 + 3] * scale);
    const float y2 = rintf(rois[k * 5 + 4] * scale);
    const int   b  = (int)r0;

    const float roi_w = fmaxf(x2 - x1 + 1.0f, 1.0f);
    const float roi_h = fmaxf(y2 - y1 + 1.0f, 1.0f);
    const float bin_w = roi_w / (float)POOL;
    const float bin_h = roi_h / (float)POOL;
    const float Hf = (float)H, Wf = (float)W;

    // Window covering all bins: identical float expressions to bin-0 start /
    // bin-6 end, so coverage holds despite fp rounding in 7*(roi_h/7).
    const int h0 = (int)clipf(y1, 0.0f, Hf);
    const int w0 = (int)clipf(x1, 0.0f, Wf);
    const int h1 = (int)clipf(ceilf(7.0f * bin_h) + y1, 0.0f, Hf);
    const int w1 = (int)clipf(ceilf(7.0f * bin_w) + x1, 0.0f, Wf);
    const int winH = h1 - h0;
    const int winW = w1 - w0;
    const int cells = winH * winW;

    const bool useLds = (winH > 0) && (winW > 0) && (cells <= CELL_CAP);

    // ---- Stage the (winW x winH x cc) tile into LDS ----
    if (useLds) {
#ifdef HAVE_TDM
        if (tid == 0) {   // one wave issues the DMA; EXEC is ignored by TDM
            const unsigned ldsAddr = (unsigned)(size_t)&sm[0];
            const unsigned long long gAddr =
                (unsigned long long)(size_t)(x + ((((size_t)b * C + c0) * H + h0) * W + w0));

            // D# group 0 (CDNA5 ISA):
            //   [1:0] count=1 | [63:32] lds_addr | [120:64] global_addr | [127:126] type=2
            u32x4 g0;
            g0.x = 1u;
            g0.y = ldsAddr;
            g0.z = (unsigned)(gAddr & 0xFFFFFFFFull);
            g0.w = (unsigned)((gAddr >> 32) & 0x01FFFFFFull) | 0x80000000u;

            // D# group 1 (§8.4): data_size=2 (4B), tensor_dim0/1 = window dims
            // (all reads in-bounds by construction), tile_dim0/1/2 = window+chan,
            // stride0 = W rows, stride1 = H*W channels. No padding/iterate/mask.
            const unsigned td0 = (unsigned)winW, td1 = (unsigned)winH;
            const unsigned s0  = (unsigned)W,    s1  = (unsigned)(H * W);
            i32x8 g1;
            g1[0] = (int)(2u << 16);                               // data_size=4B
            g1[1] = (int)((td0 & 0xFFFFu) << 16);                  // tensor_dim0 lo
            g1[2] = (int)((td0 >> 16) | ((td1 & 0xFFFFu) << 16));  // td0 hi | td1 lo
            g1[3] = (int)((td1 >> 16) | ((unsigned)winW << 16));   // td1 hi | tile_dim0
            g1[4] = (int)((unsigned)winH | ((unsigned)cc << 16));  // tile_dim1 | tile_dim2
            g1[5] = (int)s0;                                       // stride0[31:0]
            g1[6] = (int)((s1 & 0xFFFFu) << 16);                   // s0 hi(0) | s1 lo
            g1[7] = (int)(s1 >> 16);                               // s1 hi

            // D# group 2 (§8.5): tensor_dim2 = cc; dim3/stride2/tile_dim3 unused.
            i32x4 g2; g2[0] = cc; g2[1] = 0; g2[2] = 0; g2[3] = 0;
            // D# group 3 (§8.6): unused dims -> 0.
            i32x4 g3; g3[0] = 0; g3[1] = 0; g3[2] = 0; g3[3] = 0;
            // clang-23 6-arg form carries an extra int32x8 beyond the four ISA
            // descriptor groups; zero-filled (probe-verified pattern).
            i32x8 gz = {0, 0, 0, 0, 0, 0, 0, 0};

            __builtin_amdgcn_tensor_load_to_lds(g0, g1, g2, g3, gz, 0);
            __builtin_amdgcn_s_wait_tensorcnt(0);
        }
        __syncthreads();   // publish TDM-written LDS to all waves
#else
        const int total = cc * cells;
        for (int idx = tid; idx < total; idx += blockDim.x) {
            const int c = idx / cells;
            const int r = idx - c * cells;
            const int h = r / winW;
            const int w = r - h * winW;
            const float* g = x + ((((size_t)b * C + (c0 + c)) * H + (h0 + h)) * W + (w0 + w));
            async_ld_f32_to_lds(g, (unsigned)(size_t)&sm[idx]);
        }
        wait_asynccnt0();
        __syncthreads();
#endif
    }

    // ---- Max-reduce each (channel, ph, pw) bin ----
    const int outPerChunk = cc * POOL * POOL;
    for (int o = tid; o < outPerChunk; o += blockDim.x) {
        const int c   = o / (POOL * POOL);
        const int bin = o - c * (POOL * POOL);
        const int ph  = bin / POOL;
        const int pw  = bin - ph * POOL;

        const int hs = (int)clipf(floorf((float)ph * bin_h) + y1, 0.0f, Hf);
        const int he = (int)clipf(ceilf((float)(ph + 1) * bin_h) + y1, 0.0f, Hf);
        const int ws = (int)clipf(floorf((float)pw * bin_w) + x1, 0.0f, Wf);
        const int we = (int)clipf(ceilf((float)(pw + 1) * bin_w) + x1, 0.0f, Wf);

        const bool empty = (he <= hs) || (we <= ws);
        float m = -INFINITY;
        if (!empty) {
            if (useLds) {
                const float* base = &sm[c * cells];
                for (int h = hs; h < he; ++h)
                    for (int w = ws; w < we; ++w)
                        m = fmaxf(m, base[(h - h0) * winW + (w - w0)]);
            } else {
                const float* base = x + (((size_t)b * C + (c0 + c)) * (size_t)H) * W;
                for (int h = hs; h < he; ++h)
                    for (int w = ws; w < we; ++w)
                        m = fmaxf(m, base[h * W + w]);
            }
        }
        out[(((size_t)k * C + (c0 + c)) * POOL + ph) * POOL + pw] = empty ? 0.0f : m;
    }
}

extern "C" void kernel_launch(void* const* d_in, const int* in_sizes, int n_in,
                              void* d_out, int out_size, void* d_ws, size_t ws_size,
                              hipStream_t stream) {
    const float* x    = (const float*)d_in[0];
    const float* rois = (const float*)d_in[1];
    float*       out  = (float*)d_out;

    const int C = 256, H = 50, W = 50;
    const int K = in_sizes[1] / 5;
    const float scale = 0.0625f;

    dim3 grid((unsigned)K, (unsigned)((C + CCHUNK - 1) / CCHUNK));
    roipool_kernel<<<grid, 256, 0, stream>>>(x, rois, out, C, H, W, scale);
}